// MyGCN_33277406609480
// MI455X (gfx1250) — compile-verified
//
#include <hip/hip_runtime.h>
#include <math.h>

typedef __attribute__((ext_vector_type(2))) float v2f;
typedef __attribute__((ext_vector_type(8))) float v8f;

// Native GLOBAL_ATOMIC_ADD_F32 (no CAS loop). Reference is a parallel
// segment_sum, so hardware FP atomic ordering is acceptable.
__device__ __forceinline__ void atomicAddF32(float* p, float v) {
    unsafeAtomicAdd(p, v);
}

// ---------------------------------------------------------------------------
// Degree / norm kernels
// ---------------------------------------------------------------------------
__global__ __launch_bounds__(256) void deg_init_kernel(float* deg, int n) {
    int i = blockIdx.x * 256 + threadIdx.x;
    if (i < n) deg[i] = 1.0f;  // self-loop
}

__global__ __launch_bounds__(256) void deg_count_kernel(const long long* __restrict__ dst,
                                                        float* deg, int E) {
    int e = blockIdx.x * 256 + threadIdx.x;
    if (e < E) atomicAddF32(&deg[(int)dst[e]], 1.0f);
}

__global__ __launch_bounds__(256) void deg_rsqrt_kernel(float* deg, int n) {
    int i = blockIdx.x * 256 + threadIdx.x;
    if (i < n) deg[i] = rsqrtf(deg[i]);
}

// ---------------------------------------------------------------------------
// GEMM1: t1s = dinv[row] * (x[50000,256] @ W1[256,128])
// Block = 256 threads = 8 waves; block -> 16-row tile; wave w -> 16-col tile.
// x tile staged in LDS (stride padded 256->260 to avoid bank conflicts).
// Uses V_WMMA_F32_16X16X4_F32 (wave32, K-step 4).
// ---------------------------------------------------------------------------
__global__ __launch_bounds__(256) void gemm1_kernel(const float* __restrict__ x,
                                                    const float* __restrict__ W1,
                                                    const float* __restrict__ dinv,
                                                    float* __restrict__ t1s) {
    __shared__ float lds[16 * 260];
    const int row0 = blockIdx.x * 16;
    const int tid  = threadIdx.x;

    // cooperative load of 16x256 f32 tile: 1024 float4, 4 per thread
    const float4* src4 = (const float4*)(x + (size_t)row0 * 256);
    float4* dst4 = (float4*)lds;
#pragma unroll
    for (int i = 0; i < 4; ++i) {
        int idx = tid + i * 256;          // 0..1023
        int r = idx >> 6, c = idx & 63;   // row, float4-col
        dst4[r * 65 + c] = src4[r * 64 + c];
    }
    __syncthreads();

    const int wave = tid >> 5;            // 0..7 -> col tile
    const int lane = tid & 31;
    const int col0 = wave * 16;
    const int m    = lane & 15;           // row (A) / col (B) within tile
    const int koff = (lane >> 4) * 2;     // K sub-offset per lane half

    v8f acc = {0.f, 0.f, 0.f, 0.f, 0.f, 0.f, 0.f, 0.f};
#pragma unroll 8
    for (int k = 0; k < 256; k += 4) {
        v2f a, b;
        a.x = lds[m * 260 + k + koff];
        a.y = lds[m * 260 + k + koff + 1];
        b.x = W1[(size_t)(k + koff) * 128 + col0 + m];
        b.y = W1[(size_t)(k + koff + 1) * 128 + col0 + m];
        acc = __builtin_amdgcn_wmma_f32_16x16x4_f32(false, a, false, b,
                                                    (short)0, acc, false, false);
    }

    // D layout: VGPR r -> row (r | r+8 by lane half), col = col0 + (lane&15)
    const int rbase = (lane >> 4) * 8;
#pragma unroll
    for (int r = 0; r < 8; ++r) {
        int row = row0 + rbase + r;
        t1s[(size_t)row * 128 + col0 + m] = acc[r] * dinv[row];
    }
}

// ---------------------------------------------------------------------------
// Edge propagation layer 1: acc1[dst] += t1s[src]  (128 floats / edge)
// One wave per edge, each lane handles a float4 column chunk.
// acc1/t1s are 25.6MB each -> L2-resident atomics (192MB L2).
// ---------------------------------------------------------------------------
__global__ __launch_bounds__(256) void prop1_kernel(const long long* __restrict__ srcI,
                                                    const long long* __restrict__ dstI,
                                                    const float* __restrict__ t1s,
                                                    float* __restrict__ acc1, int E) {
    int t = blockIdx.x * 256 + threadIdx.x;
    int e = t >> 5;
    if (e >= E) return;
    int lane = t & 31;
    int s = (int)srcI[e];
    int d = (int)dstI[e];
    const float4 v = *(const float4*)(t1s + (size_t)s * 128 + lane * 4);
    float* p = acc1 + (size_t)d * 128 + lane * 4;
    atomicAddF32(p + 0, v.x);
    atomicAddF32(p + 1, v.y);
    atomicAddF32(p + 2, v.z);
    atomicAddF32(p + 3, v.w);
}

// h1 = relu(dinv[i]*acc1[i] + b1)
__global__ __launch_bounds__(256) void finalize1_kernel(const float* __restrict__ acc1,
                                                        const float* __restrict__ dinv,
                                                        const float* __restrict__ b1,
                                                        float* __restrict__ h1, int total) {
    int t = blockIdx.x * 256 + threadIdx.x;
    if (t >= total) return;
    int i = t >> 7, c = t & 127;
    float v = dinv[i] * acc1[t] + b1[c];
    h1[t] = v > 0.f ? v : 0.f;
}

// ---------------------------------------------------------------------------
// GEMM2: t2s = dinv[row] * (h1[50000,128] @ W2[128,40])
// Block = 128 threads = 4 waves; waves 0..2 cover col tiles (48 cols, 40 used).
// ---------------------------------------------------------------------------
__global__ __launch_bounds__(128) void gemm2_kernel(const float* __restrict__ h1,
                                                    const float* __restrict__ W2,
                                                    const float* __restrict__ dinv,
                                                    float* __restrict__ t2s) {
    __shared__ float lds[16 * 132];
    const int row0 = blockIdx.x * 16;
    const int tid  = threadIdx.x;

    const float4* src4 = (const float4*)(h1 + (size_t)row0 * 128);
    float4* dst4 = (float4*)lds;
#pragma unroll
    for (int i = 0; i < 4; ++i) {
        int idx = tid + i * 128;          // 0..511
        int r = idx >> 5, c = idx & 31;
        dst4[r * 33 + c] = src4[r * 32 + c];
    }
    __syncthreads();

    const int wave = tid >> 5;
    const int lane = tid & 31;
    if (wave >= 3) return;                // uniform per-wave exit; EXEC stays full
    const int col0 = wave * 16;
    const int m    = lane & 15;
    const int koff = (lane >> 4) * 2;
    const int cb   = col0 + m;
    const bool cok = (cb < 40);

    v8f acc = {0.f, 0.f, 0.f, 0.f, 0.f, 0.f, 0.f, 0.f};
#pragma unroll 8
    for (int k = 0; k < 128; k += 4) {
        v2f a, b;
        a.x = lds[m * 132 + k + koff];
        a.y = lds[m * 132 + k + koff + 1];
        b.x = cok ? W2[(size_t)(k + koff) * 40 + cb] : 0.f;
        b.y = cok ? W2[(size_t)(k + koff + 1) * 40 + cb] : 0.f;
        acc = __builtin_amdgcn_wmma_f32_16x16x4_f32(false, a, false, b,
                                                    (short)0, acc, false, false);
    }

    if (cok) {
        const int rbase = (lane >> 4) * 8;
#pragma unroll
        for (int r = 0; r < 8; ++r) {
            int row = row0 + rbase + r;
            t2s[(size_t)row * 40 + cb] = acc[r] * dinv[row];
        }
    }
}

// Edge propagation layer 2: acc2[dst][c] += t2s[src][c], c<40 (64 slots/edge)
__global__ __launch_bounds__(256) void prop2_kernel(const long long* __restrict__ srcI,
                                                    const long long* __restrict__ dstI,
                                                    const float* __restrict__ t2s,
                                                    float* __restrict__ acc2, int E) {
    int t = blockIdx.x * 256 + threadIdx.x;
    int e = t >> 6;
    int c = t & 63;
    if (e >= E || c >= 40) return;
    int s = (int)srcI[e];
    int d = (int)dstI[e];
    atomicAddF32(&acc2[(size_t)d * 40 + c], t2s[(size_t)s * 40 + c]);
}

// out = log_softmax(dinv[i]*acc2[i] + b2); one wave per node (wave32)
__global__ __launch_bounds__(256) void final_kernel(const float* __restrict__ acc2,
                                                    const float* __restrict__ dinv,
                                                    const float* __restrict__ b2,
                                                    float* __restrict__ out, int n) {
    int wave = threadIdx.x >> 5;
    int lane = threadIdx.x & 31;
    int node = blockIdx.x * 8 + wave;
    if (node >= n) return;
    float di = dinv[node];
    const float* row = acc2 + (size_t)node * 40;
    float v0 = di * row[lane] + b2[lane];
    float v1 = (lane < 8) ? (di * row[32 + lane] + b2[32 + lane]) : -INFINITY;
    float mx = fmaxf(v0, v1);
#pragma unroll
    for (int off = 16; off; off >>= 1) mx = fmaxf(mx, __shfl_xor(mx, off, 32));
    float s = expf(v0 - mx) + ((lane < 8) ? expf(v1 - mx) : 0.f);
#pragma unroll
    for (int off = 16; off; off >>= 1) s += __shfl_xor(s, off, 32);
    float lse = logf(s);
    out[(size_t)node * 40 + lane] = v0 - mx - lse;
    if (lane < 8) out[(size_t)node * 40 + 32 + lane] = v1 - mx - lse;
}

// ---------------------------------------------------------------------------
extern "C" void kernel_launch(void* const* d_in, const int* in_sizes, int n_in,
                              void* d_out, int out_size, void* d_ws, size_t ws_size,
                              hipStream_t stream) {
    const float*     x  = (const float*)d_in[0];
    const long long* ei = (const long long*)d_in[1];
    const float*     W1 = (const float*)d_in[2];
    const float*     b1 = (const float*)d_in[3];
    const float*     W2 = (const float*)d_in[4];
    const float*     b2 = (const float*)d_in[5];
    float* out = (float*)d_out;

    const int n = in_sizes[0] / 256;       // 50000 (multiple of 16)
    const int E = in_sizes[1] / 2;         // 1600000
    const long long* srcI = ei;
    const long long* dstI = ei + E;

    // workspace layout (floats)
    float* ws   = (float*)d_ws;
    float* dinv = ws;                            // n        (pad to 50048)
    float* t1s  = ws + 50048;                    // n*128
    float* acc1 = t1s + (size_t)n * 128;         // n*128
    float* t2s  = acc1 + (size_t)n * 128;        // n*40
    float* acc2 = t2s + (size_t)n * 40;          // n*40
    float* h1   = t1s;                           // reuse t1s after prop1

    const int rowTiles = n / 16;                 // 3125

    deg_init_kernel<<<(n + 255) / 256, 256, 0, stream>>>(dinv, n);
    deg_count_kernel<<<(E + 255) / 256, 256, 0, stream>>>(dstI, dinv, E);
    deg_rsqrt_kernel<<<(n + 255) / 256, 256, 0, stream>>>(dinv, n);

    gemm1_kernel<<<rowTiles, 256, 0, stream>>>(x, W1, dinv, t1s);

    hipMemcpyAsync(acc1, t1s, (size_t)n * 128 * sizeof(float),
                   hipMemcpyDeviceToDevice, stream);             // self-loop init
    {
        long long threads = (long long)E * 32;
        prop1_kernel<<<(int)((threads + 255) / 256), 256, 0, stream>>>(srcI, dstI, t1s, acc1, E);
    }
    finalize1_kernel<<<(n * 128 + 255) / 256, 256, 0, stream>>>(acc1, dinv, b1, h1, n * 128);

    gemm2_kernel<<<rowTiles, 128, 0, stream>>>(h1, W2, dinv, t2s);

    hipMemcpyAsync(acc2, t2s, (size_t)n * 40 * sizeof(float),
                   hipMemcpyDeviceToDevice, stream);             // self-loop init
    {
        long long threads = (long long)E * 64;
        prop2_kernel<<<(int)((threads + 255) / 256), 256, 0, stream>>>(srcI, dstI, t2s, acc2, E);
    }
    final_kernel<<<(n + 7) / 8, 256, 0, stream>>>(acc2, dinv, b2, out, n);
}